// ChannelLatencySeq2Value_51161650430181
// MI455X (gfx1250) — compile-verified
//
#include <hip/hip_runtime.h>
#include <math.h>

#define C_CH 256
#define T_LEN 4096
#define B_SZ 8
#define OPK 6

typedef __attribute__((ext_vector_type(2))) float v2f;
typedef __attribute__((ext_vector_type(8))) float v8f;

// ---------------------------------------------------------------------------
// Kernel 1: fused (effective depthwise conv) + exact parallel LIF scan.
// One block per (b, c). 256 threads, each owns a 16-timestep chunk of T=4096.
// ---------------------------------------------------------------------------
__global__ __launch_bounds__(256) void lif_kernel(
    const float* __restrict__ x,
    const float* __restrict__ w3, const float* __restrict__ b3,
    const float* __restrict__ w5, const float* __restrict__ b5,
    const float* __restrict__ w9, const float* __restrict__ b9,
    const float* __restrict__ w_red, const float* __restrict__ b_red,
    const float* __restrict__ latency_scale,
    float* __restrict__ out_true, float* __restrict__ out_act)
{
    __shared__ float s_w[6 * 9];
    __shared__ int   s_ic[6];
    __shared__ int   s_ns;
    __shared__ float s_bias;
    __shared__ float s_m[256];
    __shared__ float s_bv[256];
    __shared__ int   s_red[256];

    const int tid = threadIdx.x;
    const int b = blockIdx.x / C_CH;
    const int c = blockIdx.x % C_CH;

    if (tid < 54) s_w[tid] = 0.0f;
    __syncthreads();

    // Build effective 9-tap kernels (per x-channel slot) for drive channel c.
    if (tid == 0) {
        int icl[6];
        int ns = 0;
        float bc = b_red[c];
        for (int j = 0; j < 18; ++j) {
            int m = 18 * c + j;
            int l, K;
            const float* wseg;
            const float* bseg;
            if (m < 6 * C_CH)       { l = m;              K = 3; wseg = w3; bseg = b3; }
            else if (m < 12 * C_CH) { l = m - 6 * C_CH;   K = 5; wseg = w5; bseg = b5; }
            else                    { l = m - 12 * C_CH;  K = 9; wseg = w9; bseg = b9; }
            const int off = 4 - (K - 1) / 2;       // center-align inside 9 taps
            const int ic  = l / OPK;               // x channel feeding cat channel m
            const float wr = w_red[c * 18 + j];
            int slot = -1;
            for (int q = 0; q < ns; ++q) if (icl[q] == ic) slot = q;
            if (slot < 0) { slot = ns; icl[ns] = ic; s_ic[ns] = ic; ++ns; }
            for (int k = 0; k < K; ++k) s_w[slot * 9 + off + k] += wr * wseg[l * K + k];
            bc += wr * bseg[l];
        }
        for (int q = ns; q < 6; ++q) s_ic[q] = 0;
        s_ns = ns;
        s_bias = bc;
    }
    __syncthreads();

    const int t0 = tid * 16;
    const float A1  = 0.81873075307798185867f;   // exp(-1/5)
    const float OMA = 1.0f - A1;
    const float A16 = 0.040762203978366215f;     // exp(-16/5) = A1^16

    float acc[16];
    #pragma unroll
    for (int i = 0; i < 16; ++i) acc[i] = s_bias;

    const int ns = s_ns;
    const size_t xbase = (size_t)b * C_CH * T_LEN;
    for (int slot = 0; slot < ns; ++slot) {
        const float* xs = x + xbase + (size_t)s_ic[slot] * T_LEN;
        float wr[9];
        #pragma unroll
        for (int e = 0; e < 9; ++e) wr[e] = s_w[slot * 9 + e];
        float win[24];
        #pragma unroll
        for (int e = 0; e < 24; ++e) {
            int t = t0 - 4 + e;
            win[e] = (t >= 0 && t < T_LEN) ? xs[t] : 0.0f;
        }
        #pragma unroll
        for (int i = 0; i < 16; ++i) {
            float s = acc[i];
            #pragma unroll
            for (int e = 0; e < 9; ++e) s = fmaf(wr[e], win[i + e], s);
            acc[i] = s;
        }
    }

    // Local LIF solution from zero state; keep V_loc in acc[].
    float V = 0.0f;
    #pragma unroll
    for (int i = 0; i < 16; ++i) { V = A1 * V + OMA * acc[i]; acc[i] = V; }

    // Affine-map scan across the 256 chunks: f_i(V) = A16*V + end_i.
    float mm = A16, bb = V;
    s_m[tid] = mm; s_bv[tid] = bb;
    __syncthreads();
    for (int ofs = 1; ofs < 256; ofs <<= 1) {
        float pm = 1.0f, pb = 0.0f;
        if (tid >= ofs) { pm = s_m[tid - ofs]; pb = s_bv[tid - ofs]; }
        __syncthreads();
        if (tid >= ofs) {
            bb = fmaf(mm, pb, bb);   // later ∘ earlier
            mm = mm * pm;
            s_m[tid] = mm; s_bv[tid] = bb;
        }
        __syncthreads();
    }
    const float Vin = (tid == 0) ? 0.0f : s_bv[tid - 1];

    // Exact reconstruction: V_t = V_loc[i] + A1^(i+1) * Vin; first threshold crossing.
    int firstT = T_LEN;
    float p = A1;
    #pragma unroll
    for (int i = 0; i < 16; ++i) {
        float Vi = fmaf(p, Vin, acc[i]);
        if (Vi >= 1.0f && firstT == T_LEN) firstT = t0 + i;
        p *= A1;
    }
    s_red[tid] = firstT;
    __syncthreads();
    for (int ofs = 128; ofs > 0; ofs >>= 1) {
        if (tid < ofs) { int o = s_red[tid + ofs]; if (o < s_red[tid]) s_red[tid] = o; }
        __syncthreads();
    }
    if (tid == 0) {
        float lat = (float)s_red[0];                 // == T when never fired
        float sc = fmaxf(latency_scale[0], 0.001f);
        out_true[b * C_CH + c] = lat;
        out_act[b * C_CH + c]  = expf(-lat / sc);
    }
}

// ---------------------------------------------------------------------------
// Kernel 2: head — act@og^T + bias -> relu(@W1+b1) -> @W2+b2 -> softplus/clamp
// One workgroup, 16 waves, V_WMMA_F32_16X16X4_F32 (M padded 8->16).
// ---------------------------------------------------------------------------
__global__ __launch_bounds__(512) void head_kernel(
    const float* __restrict__ act,   // (8,256)
    const float* __restrict__ og,    // (256,256) row-major [n][c]
    const float* __restrict__ bias,  // (256)
    const float* __restrict__ W1,    // (256,128) row-major [in][out]
    const float* __restrict__ b1,    // (128)
    const float* __restrict__ W2,    // (128,256)
    const float* __restrict__ b2,    // (256)
    float* __restrict__ pred)        // (8,256)
{
    __shared__ float s_act[16 * 256];
    __shared__ float s_mix[16 * 256];
    __shared__ float s_h[16 * 128];

    const int tid = threadIdx.x;
    for (int idx = tid; idx < 16 * 256; idx += 512) {
        int r = idx >> 8;
        s_act[idx] = (r < 8) ? act[idx] : 0.0f;     // zero-pad rows 8..15
    }
    __syncthreads();

    const int lane = tid & 31;
    const int w    = tid >> 5;
    const int l15  = lane & 15;
    const int hi   = lane >> 4;       // 0: K0/K1 halves, 1: K2/K3 halves

    // ---- Stage 1: mixed = act @ og^T + bias  (16 waves x 16 cols, K=256) ----
    {
        const int col = w * 16 + l15;
        v8f d = {};
        for (int k = 0; k < 64; ++k) {
            const int kb = 4 * k + 2 * hi;
            v2f a, bf;
            a.x  = s_act[l15 * 256 + kb];
            a.y  = s_act[l15 * 256 + kb + 1];
            bf.x = og[col * 256 + kb];            // B[kk][n] = og[n][kglob]
            bf.y = og[col * 256 + kb + 1];
            d = __builtin_amdgcn_wmma_f32_16x16x4_f32(false, a, false, bf,
                                                      (short)0, d, false, false);
        }
        const float bv = bias[col];
        #pragma unroll
        for (int i = 0; i < 8; ++i) s_mix[(i + 8 * hi) * 256 + col] = d[i] + bv;
    }
    __syncthreads();

    // ---- Stage 2: h = relu(mixed @ W1 + b1)  (waves 0..7, K=256, N=128) ----
    if (w < 8) {
        const int col = w * 16 + l15;
        v8f d = {};
        for (int k = 0; k < 64; ++k) {
            const int kb = 4 * k + 2 * hi;
            v2f a, bf;
            a.x  = s_mix[l15 * 256 + kb];
            a.y  = s_mix[l15 * 256 + kb + 1];
            bf.x = W1[kb * 128 + col];
            bf.y = W1[(kb + 1) * 128 + col];
            d = __builtin_amdgcn_wmma_f32_16x16x4_f32(false, a, false, bf,
                                                      (short)0, d, false, false);
        }
        const float bv = b1[col];
        #pragma unroll
        for (int i = 0; i < 8; ++i)
            s_h[(i + 8 * hi) * 128 + col] = fmaxf(d[i] + bv, 0.0f);
    }
    __syncthreads();

    // ---- Stage 3: raw = h @ W2 + b2 -> softplus -> clamp [0, T] ----
    {
        const int col = w * 16 + l15;
        v8f d = {};
        for (int k = 0; k < 32; ++k) {
            const int kb = 4 * k + 2 * hi;
            v2f a, bf;
            a.x  = s_h[l15 * 128 + kb];
            a.y  = s_h[l15 * 128 + kb + 1];
            bf.x = W2[kb * 256 + col];
            bf.y = W2[(kb + 1) * 256 + col];
            d = __builtin_amdgcn_wmma_f32_16x16x4_f32(false, a, false, bf,
                                                      (short)0, d, false, false);
        }
        const float bv = b2[col];
        if (hi == 0) {                 // rows 0..7 live in the hi==0 half
            #pragma unroll
            for (int i = 0; i < 8; ++i) {
                float raw = d[i] + bv;
                float sp  = (raw > 20.0f) ? raw : log1pf(expf(raw));
                pred[i * 256 + col] = fminf(fmaxf(sp, 0.0f), (float)T_LEN);
            }
        }
    }
}

// ---------------------------------------------------------------------------
extern "C" void kernel_launch(void* const* d_in, const int* in_sizes, int n_in,
                              void* d_out, int out_size, void* d_ws, size_t ws_size,
                              hipStream_t stream)
{
    (void)in_sizes; (void)n_in; (void)d_ws; (void)ws_size; (void)out_size;

    const float* x     = (const float*)d_in[0];
    const float* w3    = (const float*)d_in[1];
    const float* b3    = (const float*)d_in[2];
    const float* w5    = (const float*)d_in[3];
    const float* b5    = (const float*)d_in[4];
    const float* w9    = (const float*)d_in[5];
    const float* b9    = (const float*)d_in[6];
    const float* w_red = (const float*)d_in[7];
    const float* b_red = (const float*)d_in[8];
    const float* ls    = (const float*)d_in[9];
    const float* og    = (const float*)d_in[10];
    const float* bias  = (const float*)d_in[11];
    const float* W1    = (const float*)d_in[12];
    const float* b1    = (const float*)d_in[13];
    const float* W2    = (const float*)d_in[14];
    const float* b2    = (const float*)d_in[15];

    float* pred     = (float*)d_out;                    // (8,256)
    float* out_true = (float*)d_out + B_SZ * C_CH;      // (8,256)
    float* out_act  = (float*)d_out + 2 * B_SZ * C_CH;  // (8,256)

    lif_kernel<<<B_SZ * C_CH, 256, 0, stream>>>(
        x, w3, b3, w5, b5, w9, b9, w_red, b_red, ls, out_true, out_act);

    head_kernel<<<1, 512, 0, stream>>>(
        out_act, og, bias, W1, b1, W2, b2, pred);
}